// MoleGNN_54107997995253
// MI455X (gfx1250) — compile-verified
//
#include <hip/hip_runtime.h>

#define GDIM 128
#define NGRAPH 256
#define NEG_SLOPE 0.2f

typedef float v2f __attribute__((ext_vector_type(2)));
typedef float v8f __attribute__((ext_vector_type(8)));

// ---------------- order-preserving float <-> uint encoding (for atomicMax) ----
__device__ __forceinline__ unsigned enc_f32(float f) {
    unsigned b = __float_as_uint(f);
    return (b & 0x80000000u) ? ~b : (b | 0x80000000u);
}
__device__ __forceinline__ float dec_f32(unsigned u) {
    return __uint_as_float((u & 0x80000000u) ? (u ^ 0x80000000u) : ~u);
}

// ---------------- GEMM: H[n,128] = X[n,128] * W[128,128] via WMMA f32 16x16x4 --
__global__ void __launch_bounds__(128)
gat_gemm_wmma(const float* __restrict__ X, const float* __restrict__ W,
              float* __restrict__ H, int nTiles) {
    extern __shared__ float sW[];               // 128*128 fp32 = 64 KB
    const int tid = threadIdx.x;

    // cooperative load of W into LDS, float4 per thread x 32
    const float4* W4 = (const float4*)W;
    float4*       S4 = (float4*)sW;
#pragma unroll
    for (int t = 0; t < 32; ++t) S4[t * 128 + tid] = W4[t * 128 + tid];
    __syncthreads();

    const int wid  = tid >> 5;
    const int lane = tid & 31;
    const int hi   = lane >> 4;                 // 0 or 1 (lane half)
    const int col  = lane & 15;
    const int tile = blockIdx.x * 4 + wid;      // 4 waves / block
    if (tile >= nTiles) return;                 // wave-uniform: EXEC stays all-1s

    const long   row0 = (long)tile * 16;
    const float* Xr   = X + (row0 + col) * GDIM;

    v8f c[8];
#pragma unroll
    for (int n = 0; n < 8; ++n) c[n] = (v8f)0.0f;

    for (int kk = 0; kk < GDIM; kk += 4) {
        const int ka = kk + 2 * hi;             // this lane-half's K pair
        v2f a = *(const v2f*)(Xr + ka);         // A: 16x4, row=col(lane), K=ka..ka+1
#pragma unroll
        for (int n = 0; n < 8; ++n) {
            v2f b;
            b.x = sW[(ka + 0) * GDIM + n * 16 + col];   // B: 4x16
            b.y = sW[(ka + 1) * GDIM + n * 16 + col];
            c[n] = __builtin_amdgcn_wmma_f32_16x16x4_f32(
                       false, a, false, b, (short)0, c[n], false, false);
        }
    }

    // C/D layout: vgpr v, lanes0-15 -> M=v, lanes16-31 -> M=v+8; N=lane&15
    float* Hr = H + row0 * GDIM;
#pragma unroll
    for (int n = 0; n < 8; ++n)
#pragma unroll
        for (int v = 0; v < 8; ++v)
            Hr[(v + 8 * hi) * GDIM + n * 16 + col] = c[n][v];
}

// ---------------- per-node attention scores: s = h . a ------------------------
__global__ void __launch_bounds__(256)
gat_node_scores(const float* __restrict__ H, const float* __restrict__ a_src,
                const float* __restrict__ a_dst, float* __restrict__ s_src,
                float* __restrict__ s_dst, int n) {
    const int lane = threadIdx.x & 31;
    const int node = blockIdx.x * 8 + (threadIdx.x >> 5);
    if (node >= n) return;
    const float4 hv = *(const float4*)(H + (long)node * GDIM + lane * 4);
    const float4 as = *(const float4*)(a_src + lane * 4);
    const float4 ad = *(const float4*)(a_dst + lane * 4);
    float ps = hv.x * as.x + hv.y * as.y + hv.z * as.z + hv.w * as.w;
    float pd = hv.x * ad.x + hv.y * ad.y + hv.z * ad.z + hv.w * ad.w;
#pragma unroll
    for (int off = 16; off > 0; off >>= 1) {
        ps += __shfl_xor(ps, off, 32);
        pd += __shfl_xor(pd, off, 32);
    }
    if (lane == 0) { s_src[node] = ps; s_dst[node] = pd; }
}

// ---------------- per-edge segment max of logits ------------------------------
__global__ void __launch_bounds__(256)
gat_edge_max(const int* __restrict__ ei, int E, int n,
             const float* __restrict__ s_src, const float* __restrict__ s_dst,
             unsigned* __restrict__ m_enc) {
    const int e  = blockIdx.x * blockDim.x + threadIdx.x;
    const int NE = E + n;
    if (e >= NE) return;
    int i, j;
    if (e < E) { i = ei[e]; j = ei[E + e]; } else { i = j = e - E; }
    const float s     = s_src[i] + s_dst[j];
    const float logit = (s > 0.f) ? s : NEG_SLOPE * s;
    atomicMax(m_enc + j, enc_f32(logit));
}

// ---------------- per-edge: w=exp(logit-max); denom += w; acc[dst] += w*h[src] -
__global__ void __launch_bounds__(256)
gat_edge_aggr(const int* __restrict__ ei, int E, int n,
              const float* __restrict__ s_src, const float* __restrict__ s_dst,
              const unsigned* __restrict__ m_enc, const float* __restrict__ H,
              float* __restrict__ denom, float* __restrict__ acc) {
    const int lane = threadIdx.x & 31;
    const int e    = blockIdx.x * 8 + (threadIdx.x >> 5);   // one wave per edge
    const int NE   = E + n;
    if (e >= NE) return;
    int i, j;
    if (e < E) { i = ei[e]; j = ei[E + e]; } else { i = j = e - E; }
    const float s     = s_src[i] + s_dst[j];
    const float logit = (s > 0.f) ? s : NEG_SLOPE * s;
    const float w     = __expf(logit - dec_f32(m_enc[j]));
    if (lane == 0) atomicAdd(denom + j, w);
    const float4 hv = *(const float4*)(H + (long)i * GDIM + lane * 4);
    float* ap = acc + (long)j * GDIM + lane * 4;
    atomicAdd(ap + 0, w * hv.x);
    atomicAdd(ap + 1, w * hv.y);
    atomicAdd(ap + 2, w * hv.z);
    atomicAdd(ap + 3, w * hv.w);
}

// ---------------- node finalize: acc = acc/denom + bias (in place) ------------
__global__ void __launch_bounds__(256)
gat_finalize(float* __restrict__ acc, const float* __restrict__ denom,
             const float* __restrict__ bias, int n) {
    const int lane = threadIdx.x & 31;
    const int node = blockIdx.x * 8 + (threadIdx.x >> 5);
    if (node >= n) return;
    const float  inv = 1.0f / denom[node];
    const float4 b4  = *(const float4*)(bias + lane * 4);
    float4* p = (float4*)(acc + (long)node * GDIM + lane * 4);
    float4  v = *p;
    v.x = v.x * inv + b4.x;
    v.y = v.y * inv + b4.y;
    v.z = v.z * inv + b4.z;
    v.w = v.w * inv + b4.w;
    *p = v;
}

// ---------------- global mean pool: accumulate ---------------------------------
__global__ void __launch_bounds__(256)
gat_pool_accum(const float* __restrict__ h, const int* __restrict__ batch, int n,
               float* __restrict__ psum, float* __restrict__ pcount) {
    const int lane = threadIdx.x & 31;
    const int node = blockIdx.x * 8 + (threadIdx.x >> 5);
    if (node >= n) return;
    const int    g  = batch[node];
    const float4 hv = *(const float4*)(h + (long)node * GDIM + lane * 4);
    float* pp = psum + (long)g * GDIM + lane * 4;
    atomicAdd(pp + 0, hv.x);
    atomicAdd(pp + 1, hv.y);
    atomicAdd(pp + 2, hv.z);
    atomicAdd(pp + 3, hv.w);
    if (lane == 0) atomicAdd(pcount + g, 1.0f);
}

// ---------------- pool finalize: out = psum / max(count,1) --------------------
__global__ void __launch_bounds__(256)
gat_pool_final(const float* __restrict__ psum, const float* __restrict__ pcount,
               float* __restrict__ out) {
    const int idx = blockIdx.x * blockDim.x + threadIdx.x;   // 256*128 total
    const float cnt = pcount[idx >> 7];
    out[idx] = psum[idx] / fmaxf(cnt, 1.0f);
}

extern "C" void kernel_launch(void* const* d_in, const int* in_sizes, int n_in,
                              void* d_out, int out_size, void* d_ws, size_t ws_size,
                              hipStream_t stream) {
    const float* x     = (const float*)d_in[0];
    const int*   ei    = (const int*)d_in[1];
    const int*   batch = (const int*)d_in[2];
    const float* W[3]    = {(const float*)d_in[3], (const float*)d_in[7],  (const float*)d_in[11]};
    const float* asrc[3] = {(const float*)d_in[4], (const float*)d_in[8],  (const float*)d_in[12]};
    const float* adst[3] = {(const float*)d_in[5], (const float*)d_in[9],  (const float*)d_in[13]};
    const float* bias[3] = {(const float*)d_in[6], (const float*)d_in[10], (const float*)d_in[14]};

    const int n  = in_sizes[0] / GDIM;   // 100000
    const int E  = in_sizes[1] / 2;      // 800000
    const int NE = E + n;

    // -------- workspace layout --------
    char*  wsb = (char*)d_ws;
    size_t off = 0;
    auto take = [&](size_t bytes) -> void* {
        void* p = wsb + off;
        off += (bytes + 255) & ~(size_t)255;
        return p;
    };
    const size_t szH = (size_t)n * GDIM * sizeof(float);
    float*    h      = (float*)   take(szH);
    float*    acc    = (float*)   take(szH);
    float*    s_src  = (float*)   take((size_t)n * sizeof(float));
    float*    s_dst  = (float*)   take((size_t)n * sizeof(float));
    unsigned* m_enc  = (unsigned*)take((size_t)n * sizeof(unsigned));
    float*    denom  = (float*)   take((size_t)n * sizeof(float));
    float*    psum   = (float*)   take((size_t)NGRAPH * GDIM * sizeof(float));
    float*    pcount = (float*)   take((size_t)NGRAPH * sizeof(float));
    (void)ws_size; (void)n_in; (void)out_size;

    const int nTiles     = (n + 15) / 16;          // 6250 (n divisible by 16)
    const int gemmBlocks = (nTiles + 3) / 4;
    const int nodeBlocks = (n + 7) / 8;
    const int edgeTBlk   = (NE + 255) / 256;
    const int edgeWBlk   = (NE + 7) / 8;

    const float* xin = x;
    for (int l = 0; l < 3; ++l) {
        gat_gemm_wmma<<<gemmBlocks, 128, GDIM * GDIM * sizeof(float), stream>>>(
            xin, W[l], h, nTiles);
        gat_node_scores<<<nodeBlocks, 256, 0, stream>>>(h, asrc[l], adst[l],
                                                        s_src, s_dst, n);
        hipMemsetAsync(m_enc, 0, (size_t)n * sizeof(unsigned), stream);
        hipMemsetAsync(denom, 0, (size_t)n * sizeof(float), stream);
        hipMemsetAsync(acc,   0, szH, stream);     // safe: GEMM already consumed it
        gat_edge_max<<<edgeTBlk, 256, 0, stream>>>(ei, E, n, s_src, s_dst, m_enc);
        gat_edge_aggr<<<edgeWBlk, 256, 0, stream>>>(ei, E, n, s_src, s_dst,
                                                    m_enc, h, denom, acc);
        gat_finalize<<<nodeBlocks, 256, 0, stream>>>(acc, denom, bias[l], n);
        xin = acc;                                  // next layer input (in place)
    }

    hipMemsetAsync(psum,   0, (size_t)NGRAPH * GDIM * sizeof(float), stream);
    hipMemsetAsync(pcount, 0, (size_t)NGRAPH * sizeof(float), stream);
    gat_pool_accum<<<nodeBlocks, 256, 0, stream>>>(acc, batch, n, psum, pcount);
    gat_pool_final<<<(NGRAPH * GDIM) / 256, 256, 0, stream>>>(psum, pcount,
                                                              (float*)d_out);
}